// BatchFeatureDecorr_37855841747396
// MI455X (gfx1250) — compile-verified
//
#include <hip/hip_runtime.h>

// ---------------------------------------------------------------------------
// BatchFeatureDecorr for MI455X (gfx1250, wave32).
// Big GEMMs (Gram = X X^T and out = W2 X + b) via V_WMMA_F32_16X16X4_F32.
// ---------------------------------------------------------------------------

typedef __attribute__((ext_vector_type(2))) float v2f;
typedef __attribute__((ext_vector_type(8))) float v8f;

#define GDIM      64            // groups
#define HW        3136          // 56*56
#define P_COUNT   256           // N * (C/G) = 64*4
#define HWT       49            // 3136 / 64
#define TILES     (P_COUNT * HWT)   // 12544 tiles of 64x64
#define TILE_HW   64
#define M_TOTAL   802816.0f     // P_COUNT * HW
#define EPSV      1e-5f
#define N_ITER    10
#define NWG_A     256
#define LSTR      65            // LDS row stride (pad to dodge 64-bank conflicts)

__device__ __forceinline__ v8f wmma4(v2f a, v2f b, v8f c) {
    // 8-arg pattern for 16x16x4/32 shapes: (neg_a, A, neg_b, B, c_mod, C, reuse_a, reuse_b)
    return __builtin_amdgcn_wmma_f32_16x16x4_f32(false, a, false, b, (short)0, c,
                                                 false, false);
}

// ---------------------------------------------------------------------------
// Kernel 1: per-workgroup partial Gram (64x64) and per-group partial sums.
// ---------------------------------------------------------------------------
__global__ __launch_bounds__(256)
void gram_kernel(const float* __restrict__ x,
                 float* __restrict__ gramP,   // [NWG_A][64*64]
                 float* __restrict__ sumP)    // [NWG_A][64]
{
    __shared__ float Xt[GDIM][LSTR];
    __shared__ float ssum[256];

    const int t    = threadIdx.x;
    const int lane = t & 31;
    const int wave = t >> 5;                 // 0..7

    // Each wave owns Gram tiles 2w and 2w+1 (same i block, two j blocks).
    const int t0 = 2 * wave;
    const int i0 = t0 >> 2;
    const int j0 = t0 & 3;
    const int j1 = (t0 + 1) & 3;

    // WMMA f32 16x16x4 fragment decode (A and B use the same lane pattern):
    const int frow = lane & 15;              // matrix row (A) / col (B) in tile
    const int koff = (lane >> 4) * 2;        // K base for this lane half

    // Coalesced tile loads: 16 lanes * float4 = 256B rows.
    const int row0 = t >> 4;                 // 0..15
    const int c4   = (t & 15) * 4;           // 0..60

    v8f acc0 = {0.f,0.f,0.f,0.f,0.f,0.f,0.f,0.f};
    v8f acc1 = {0.f,0.f,0.f,0.f,0.f,0.f,0.f,0.f};

    // per-thread partial sum: group sg, hw quarter sq
    const int sg = t & 63, sq = t >> 6;
    float srow = 0.f;

    for (int tile = blockIdx.x; tile < TILES; tile += gridDim.x) {
        const int p   = tile / HWT;
        const int hw0 = (tile - p * HWT) * TILE_HW;
        const float* base = x + (size_t)p * GDIM * HW + hw0;

        #pragma unroll
        for (int k = 0; k < 4; ++k) {
            const int r = row0 + k * 16;
            float4 v = *(const float4*)(base + (size_t)r * HW + c4);
            Xt[r][c4+0] = v.x; Xt[r][c4+1] = v.y;
            Xt[r][c4+2] = v.z; Xt[r][c4+3] = v.w;
        }
        __syncthreads();

        #pragma unroll
        for (int k = 0; k < 16; ++k) srow += Xt[sg][sq * 16 + k];

        #pragma unroll
        for (int kk = 0; kk < 16; ++kk) {
            v2f A, B0, B1;
            A.x  = Xt[i0*16 + frow][kk*4 + koff + 0];
            A.y  = Xt[i0*16 + frow][kk*4 + koff + 1];
            B0.x = Xt[j0*16 + frow][kk*4 + koff + 0];
            B0.y = Xt[j0*16 + frow][kk*4 + koff + 1];
            B1.x = Xt[j1*16 + frow][kk*4 + koff + 0];
            B1.y = Xt[j1*16 + frow][kk*4 + koff + 1];
            acc0 = wmma4(A, B0, acc0);
            acc1 = wmma4(A, B1, acc1);
        }
        __syncthreads();
    }

    // Store partial Gram: D element (r + 8*(lane>>4), lane&15) in VGPR r.
    float* gp = gramP + (size_t)blockIdx.x * (GDIM * GDIM);
    const int drow = 8 * (lane >> 4);
    const int dcol = lane & 15;
    #pragma unroll
    for (int r = 0; r < 8; ++r) {
        gp[(i0*16 + drow + r) * GDIM + j0*16 + dcol] = acc0[r];
        gp[(i0*16 + drow + r) * GDIM + j1*16 + dcol] = acc1[r];
    }

    ssum[t] = srow;
    __syncthreads();
    if (t < GDIM)
        sumP[(size_t)blockIdx.x * GDIM + t] =
            ssum[t] + ssum[t + 64] + ssum[t + 128] + ssum[t + 192];
}

// ---------------------------------------------------------------------------
// Kernel 2: coalesced reduction of the NWG_A partials.
// ---------------------------------------------------------------------------
__global__ __launch_bounds__(256)
void reduce_kernel(const float* __restrict__ gramP, const float* __restrict__ sumP,
                   float* __restrict__ gramF, float* __restrict__ sumF)
{
    const int bid = blockIdx.x, t = threadIdx.x;
    if (bid < 16) {
        const int idx = bid * 256 + t;            // 16*256 = 4096 entries
        float s = 0.f;
        for (int b = 0; b < NWG_A; ++b) s += gramP[(size_t)b * (GDIM*GDIM) + idx];
        gramF[idx] = s;
    } else if (t < GDIM) {
        float s = 0.f;
        for (int b = 0; b < NWG_A; ++b) s += sumP[(size_t)b * GDIM + t];
        sumF[t] = s;
    }
}

// ---------------------------------------------------------------------------
// Kernel 3 (single workgroup): cov, Newton-Schulz isqrt, W2 and b_eff.
// 64x64 matmuls are ~16 MFLOP total -> scalar FMA is fine.
// ---------------------------------------------------------------------------
__global__ __launch_bounds__(256)
void solve_kernel(const float* __restrict__ w1, const float* __restrict__ b1,
                  const float* __restrict__ gramF, const float* __restrict__ sumF,
                  float* __restrict__ W2out, float* __restrict__ beffout)
{
    __shared__ float Y[GDIM][GDIM];
    __shared__ float Z[GDIM][GDIM];
    __shared__ float Tm[GDIM][GDIM];
    __shared__ float mean[GDIM];
    __shared__ float red[256];

    const int t = threadIdx.x;
    if (t < GDIM) mean[t] = sumF[t] / M_TOTAL;
    __syncthreads();

    // cov into Y, accumulate squared Frobenius norm
    float fn = 0.f;
    #pragma unroll
    for (int e = 0; e < 16; ++e) {
        const int idx = t * 16 + e;
        const int r = idx >> 6, c = idx & 63;
        float cov = gramF[idx] / M_TOTAL - mean[r] * mean[c] + ((r == c) ? EPSV : 0.f);
        Y[r][c] = cov;
        fn += cov * cov;
    }
    red[t] = fn;
    __syncthreads();
    for (int s = 128; s > 0; s >>= 1) {
        if (t < s) red[t] += red[t + s];
        __syncthreads();
    }
    const float normA = sqrtf(red[0]);
    const float inv   = 1.0f / normA;
    __syncthreads();

    #pragma unroll
    for (int e = 0; e < 16; ++e) {
        const int idx = t * 16 + e;
        const int r = idx >> 6, c = idx & 63;
        Y[r][c] *= inv;
        Z[r][c] = (r == c) ? 1.f : 0.f;
    }
    __syncthreads();

    const int row = t >> 2;                 // 64 rows, 4 threads per row
    const int c0  = (t & 3) * 16;           // 16 columns each

    for (int it = 0; it < N_ITER; ++it) {
        // T = 0.5 * (3I - Z @ Y)
        float acc[16];
        #pragma unroll
        for (int c = 0; c < 16; ++c) acc[c] = 0.f;
        for (int k = 0; k < GDIM; ++k) {
            const float zv = Z[row][k];
            #pragma unroll
            for (int c = 0; c < 16; ++c) acc[c] += zv * Y[k][c0 + c];
        }
        #pragma unroll
        for (int c = 0; c < 16; ++c) {
            const int cc = c0 + c;
            Tm[row][cc] = 0.5f * (((row == cc) ? 3.f : 0.f) - acc[c]);
        }
        __syncthreads();

        // Ynew = Y @ T ; Znew = T @ Z   (into registers, then write back)
        float ya[16], za[16];
        #pragma unroll
        for (int c = 0; c < 16; ++c) { ya[c] = 0.f; za[c] = 0.f; }
        for (int k = 0; k < GDIM; ++k) {
            const float yv = Y[row][k];
            const float tv = Tm[row][k];
            #pragma unroll
            for (int c = 0; c < 16; ++c) {
                ya[c] += yv * Tm[k][c0 + c];
                za[c] += tv * Z[k][c0 + c];
            }
        }
        __syncthreads();
        #pragma unroll
        for (int c = 0; c < 16; ++c) {
            Y[row][c0 + c] = ya[c];
            Z[row][c0 + c] = za[c];
        }
        __syncthreads();
    }

    // W2 = w1 @ (Z / sqrt(normA)) ;  b_eff = b1 - W2 @ mean
    const float rs = rsqrtf(normA);
    float wa[16];
    #pragma unroll
    for (int c = 0; c < 16; ++c) wa[c] = 0.f;
    for (int k = 0; k < GDIM; ++k) {
        const float wv = w1[row * GDIM + k];
        #pragma unroll
        for (int c = 0; c < 16; ++c) wa[c] += wv * (Z[k][c0 + c] * rs);
    }
    float bpart = 0.f;
    #pragma unroll
    for (int c = 0; c < 16; ++c) {
        W2out[row * GDIM + c0 + c] = wa[c];
        bpart += wa[c] * mean[c0 + c];
    }
    red[t] = bpart;
    __syncthreads();
    if ((t & 3) == 0)
        beffout[row] = b1[row] - (red[t] + red[t+1] + red[t+2] + red[t+3]);
}

// ---------------------------------------------------------------------------
// Kernel 4: out = W2 @ X + b_eff, tiled 64x64, WMMA f32 16x16x4.
// ---------------------------------------------------------------------------
__global__ __launch_bounds__(256)
void apply_kernel(const float* __restrict__ x, const float* __restrict__ W2,
                  const float* __restrict__ beff, float* __restrict__ out)
{
    __shared__ float Wt[GDIM][LSTR];
    __shared__ float Xt[GDIM][LSTR];
    __shared__ float Ot[GDIM][LSTR];
    __shared__ float bsh[GDIM];

    const int t    = threadIdx.x;
    const int lane = t & 31;
    const int wave = t >> 5;

    {   // load W2 + b_eff once
        const int r = t >> 2, cc = (t & 3) * 16;
        #pragma unroll
        for (int c = 0; c < 16; ++c) Wt[r][cc + c] = W2[r * GDIM + cc + c];
        if (t < GDIM) bsh[t] = beff[t];
    }
    __syncthreads();

    const int t0 = 2 * wave;
    const int i0 = t0 >> 2;                  // output row block
    const int j0 = t0 & 3;                   // hw column blocks
    const int j1 = (t0 + 1) & 3;
    const int frow = lane & 15;
    const int koff = (lane >> 4) * 2;
    const int drow = 8 * (lane >> 4);
    const int dcol = lane & 15;
    const int row0 = t >> 4;
    const int c4   = (t & 15) * 4;

    for (int tile = blockIdx.x; tile < TILES; tile += gridDim.x) {
        const int p   = tile / HWT;
        const int hw0 = (tile - p * HWT) * TILE_HW;
        const float* base  = x   + (size_t)p * GDIM * HW + hw0;
        float*       obase = out + (size_t)p * GDIM * HW + hw0;

        #pragma unroll
        for (int k = 0; k < 4; ++k) {
            const int r = row0 + k * 16;
            float4 v = *(const float4*)(base + (size_t)r * HW + c4);
            Xt[r][c4+0] = v.x; Xt[r][c4+1] = v.y;
            Xt[r][c4+2] = v.z; Xt[r][c4+3] = v.w;
        }
        __syncthreads();

        v8f acc0 = {0.f,0.f,0.f,0.f,0.f,0.f,0.f,0.f};
        v8f acc1 = {0.f,0.f,0.f,0.f,0.f,0.f,0.f,0.f};
        #pragma unroll
        for (int kk = 0; kk < 16; ++kk) {
            v2f A, B0, B1;
            A.x  = Wt[i0*16 + frow][kk*4 + koff + 0];
            A.y  = Wt[i0*16 + frow][kk*4 + koff + 1];
            // B[k, n] = Xt[channel = kk*4+koff+v][hw col = j*16 + n]
            B0.x = Xt[kk*4 + koff + 0][j0*16 + frow];
            B0.y = Xt[kk*4 + koff + 1][j0*16 + frow];
            B1.x = Xt[kk*4 + koff + 0][j1*16 + frow];
            B1.y = Xt[kk*4 + koff + 1][j1*16 + frow];
            acc0 = wmma4(A, B0, acc0);
            acc1 = wmma4(A, B1, acc1);
        }

        #pragma unroll
        for (int r = 0; r < 8; ++r) {
            const int orow = i0*16 + drow + r;
            Ot[orow][j0*16 + dcol] = acc0[r] + bsh[orow];
            Ot[orow][j1*16 + dcol] = acc1[r] + bsh[orow];
        }
        __syncthreads();

        #pragma unroll
        for (int k = 0; k < 4; ++k) {
            const int r = row0 + k * 16;
            float4 v;
            v.x = Ot[r][c4+0]; v.y = Ot[r][c4+1];
            v.z = Ot[r][c4+2]; v.w = Ot[r][c4+3];
            *(float4*)(obase + (size_t)r * HW + c4) = v;
        }
        __syncthreads();
    }
}

// ---------------------------------------------------------------------------
// Launch
// ---------------------------------------------------------------------------
extern "C" void kernel_launch(void* const* d_in, const int* in_sizes, int n_in,
                              void* d_out, int out_size, void* d_ws, size_t ws_size,
                              hipStream_t stream) {
    const float* x  = (const float*)d_in[0];   // (64,256,56,56)
    const float* w1 = (const float*)d_in[1];   // (64,64)
    const float* b1 = (const float*)d_in[2];   // (64,1)
    float* out = (float*)d_out;

    float* ws    = (float*)d_ws;
    float* gramP = ws;                                        // NWG_A * 4096
    float* sumP  = gramP + (size_t)NWG_A * GDIM * GDIM;       // NWG_A * 64
    float* gramF = sumP  + (size_t)NWG_A * GDIM;              // 4096
    float* sumF  = gramF + GDIM * GDIM;                       // 64
    float* W2    = sumF  + GDIM;                              // 4096
    float* beff  = W2    + GDIM * GDIM;                       // 64

    gram_kernel  <<<NWG_A, 256, 0, stream>>>(x, gramP, sumP);
    reduce_kernel<<<17,    256, 0, stream>>>(gramP, sumP, gramF, sumF);
    solve_kernel <<<1,     256, 0, stream>>>(w1, b1, gramF, sumF, W2, beff);
    apply_kernel <<<1024,  256, 0, stream>>>(x, W2, beff, out);
}